// AttnResModule2D_70849780514864
// MI455X (gfx1250) — compile-verified
//
#include <hip/hip_runtime.h>
#include <math.h>

typedef __attribute__((ext_vector_type(2))) float v2f;
typedef __attribute__((ext_vector_type(4))) float v4f;
typedef __attribute__((ext_vector_type(8))) float v8f;

static constexpr int L_  = 8;
static constexpr int LP1 = 9;     // L+1 column scores per (b,s)
static constexpr int B_  = 4;
static constexpr int S_  = 2048;
static constexpr int D_  = 1024;
static constexpr int NC  = 64;    // sequence chunks for h_row scan
static constexpr int CS  = 32;    // chunk length (NC*CS == S_)
#define LN_EPS 1e-5f

// ---------------------------------------------------------------------------
// Kernel 0: wg[d] = w[d]*gamma[d];  scal[0]=sum(w*gamma), scal[1]=sum(w*beta)
// ---------------------------------------------------------------------------
__global__ __launch_bounds__(1024) void k0_prep(const float* __restrict__ w,
                                                const float* __restrict__ gamma,
                                                const float* __restrict__ beta,
                                                float* __restrict__ wg,
                                                float* __restrict__ scal) {
  __shared__ float r1[1024];
  __shared__ float r2[1024];
  const int t = threadIdx.x;          // D_ == 1024 == blockDim
  float vg = w[t] * gamma[t];
  float vb = w[t] * beta[t];
  wg[t] = vg;
  r1[t] = vg; r2[t] = vb;
  __syncthreads();
  for (int off = 512; off > 0; off >>= 1) {
    if (t < off) { r1[t] += r1[t + off]; r2[t] += r2[t + off]; }
    __syncthreads();
  }
  if (t == 0) { scal[0] = r1[0]; scal[1] = r2[0]; }
}

// ---------------------------------------------------------------------------
// Kernel 1: scores.  One wave per (b, l, 16-row s-tile).
//   score[b,s,l] = rstd * (dot(x, w*gamma) - mean*sum(w*gamma)) + sum(w*beta)
// dot(x,wg) and sum(x) computed with V_WMMA_F32_16X16X4_F32 (B col0=wg, col1=1).
// sum(x^2) via per-lane FMA + shfl_xor(16).
// A 16x4 f32 layout: lane m(<16): k0,k0+1 ; lane m+16: k0+2,k0+3  (row m = s0+m)
// B 4x16  f32 layout: lane n(<16): B[0..1, n] ; lane n+16: B[2..3, n]
// ---------------------------------------------------------------------------
__global__ __launch_bounds__(256) void k1_scores(const float* __restrict__ hist,
                                                 const float* __restrict__ cur,
                                                 const float* __restrict__ wg,
                                                 const float* __restrict__ scal,
                                                 float* __restrict__ sc) {
  const int lane = threadIdx.x & 31;
  const int wid  = blockIdx.x * 8 + (threadIdx.x >> 5);   // 0..4607
  const int l    = wid % LP1;
  const int tmp  = wid / LP1;                              // 0..511
  const int s0   = (tmp & 127) * 16;                       // S_/16 = 128 tiles
  const int b    = tmp >> 7;

  const float* X = (l < L_)
      ? (hist + (((size_t)l * B_ + b) * S_) * (size_t)D_)
      : (cur  + ((size_t)b * S_) * (size_t)D_);

  const int row = lane & 15;
  const int h   = lane >> 4;                               // k-half
  const float* xrow = X + (size_t)(s0 + row) * D_ + 2 * h;
  const float* wgp  = wg + 2 * h;
  const float swg = scal[0];
  const float swb = scal[1];

  v8f c = {0.f, 0.f, 0.f, 0.f, 0.f, 0.f, 0.f, 0.f};
  float ssq = 0.f;

  for (int k0 = 0; k0 < D_; k0 += 4) {
    v2f a = *(const v2f*)(xrow + k0);
    ssq += a.x * a.x + a.y * a.y;
    v2f g = *(const v2f*)(wgp + k0);        // lanes 0-15 same addr, 16-31 same
    v2f bv;
    bv.x = (row == 0) ? g.x : ((row == 1) ? 1.f : 0.f);
    bv.y = (row == 0) ? g.y : ((row == 1) ? 1.f : 0.f);
    c = __builtin_amdgcn_wmma_f32_16x16x4_f32(false, a, false, bv,
                                              (short)0, c, false, false);
  }
  // per-row sum of squares: lanes m and m+16 each hold half the k's of row m
  ssq += __shfl_xor(ssq, 16, 32);

  // Extract C[row,0] (dot wg) and C[row,1] (sum): M<8 -> VGPR M, lanes 0/1;
  // M>=8 -> VGPR M-8, lanes 16/17.
  float dw = 0.f, sm = 0.f;
#pragma unroll
  for (int r = 0; r < 16; ++r) {
    const int src = (r >> 3) << 4;
    float aa = __shfl(c[r & 7], src, 32);
    float bb = __shfl(c[r & 7], src + 1, 32);
    if (lane == r) { dw = aa; sm = bb; }
  }

  if (lane < 16) {
    const float mean = sm * (1.f / D_);
    const float var  = ssq * (1.f / D_) - mean * mean;
    const float rstd = rsqrtf(var + LN_EPS);
    const float score = rstd * (dw - mean * swg) + swb;
    sc[((size_t)(b * S_ + s0 + lane)) * LP1 + l] = score;
  }
}

// ---------------------------------------------------------------------------
// Kernel 2: per-batch softmax prep.  One 1024-thread block per b; thread t
// handles rows s0=2t, 2t+1.  Computes M_b (global max), q_s = exp(r_s - M),
// inclusive scan E_t, Z_t = sumExpCol_t + E_t.  Overwrites sc with normalized
// column weights p = exp(c-M)/Z; writes q (unnormalized) and invZ.
// ---------------------------------------------------------------------------
__global__ __launch_bounds__(1024) void k2_softmax(float* __restrict__ sc,
                                                   float* __restrict__ q,
                                                   float* __restrict__ invZ) {
  __shared__ float lds[1024];
  const int b = blockIdx.x;
  const int t = threadIdx.x;
  const int s0 = 2 * t;
  float* scb = sc + (size_t)b * S_ * LP1;

  float ex[2 * LP1];
  float m = -3.4e38f;
#pragma unroll
  for (int i = 0; i < 2 * LP1; ++i) {
    ex[i] = scb[(size_t)s0 * LP1 + i];    // rows s0, s0+1 are contiguous
    m = fmaxf(m, ex[i]);
  }
  lds[t] = m;
  __syncthreads();
  for (int off = 512; off > 0; off >>= 1) {
    if (t < off) lds[t] = fmaxf(lds[t], lds[t + off]);
    __syncthreads();
  }
  const float M = lds[0];
  __syncthreads();

  float se0 = 0.f, se1 = 0.f;
#pragma unroll
  for (int i = 0; i < LP1; ++i) { ex[i] = expf(ex[i] - M); se0 += ex[i]; }
#pragma unroll
  for (int i = LP1; i < 2 * LP1; ++i) { ex[i] = expf(ex[i] - M); se1 += ex[i]; }
  const float q0 = ex[LP1 - 1];
  const float q1 = ex[2 * LP1 - 1];

  // inclusive Hillis-Steele scan over per-thread pair sums
  const float ps = q0 + q1;
  lds[t] = ps;
  __syncthreads();
  for (int off = 1; off < 1024; off <<= 1) {
    float v = (t >= off) ? lds[t - off] : 0.f;
    __syncthreads();
    lds[t] += v;
    __syncthreads();
  }
  const float excl = lds[t] - ps;
  const float E0 = excl + q0;
  const float E1 = E0 + q1;
  const float iz0 = 1.f / (se0 + E0);
  const float iz1 = 1.f / (se1 + E1);

  q[(size_t)b * S_ + s0]     = q0;
  q[(size_t)b * S_ + s0 + 1] = q1;
  invZ[(size_t)b * S_ + s0]     = iz0;
  invZ[(size_t)b * S_ + s0 + 1] = iz1;
#pragma unroll
  for (int i = 0; i < LP1; ++i) {
    scb[(size_t)s0 * LP1 + i]       = ex[i] * iz0;
    scb[(size_t)(s0 + 1) * LP1 + i] = ex[LP1 + i] * iz1;
  }
}

// ---------------------------------------------------------------------------
// Kernel 3: per-chunk totals G[b][j][d] = sum_{s in chunk j} q_s * cur[b,s,d]
// grid = B*NC*2 blocks (d split in halves), 128 threads * float4.
// ---------------------------------------------------------------------------
__global__ __launch_bounds__(128) void k3_chunks(const float* __restrict__ cur,
                                                 const float* __restrict__ q,
                                                 float* __restrict__ G) {
  const int blk = blockIdx.x;
  const int dh = blk & 1;
  const int j  = (blk >> 1) & (NC - 1);
  const int b  = blk >> 7;                 // NC*2 == 128
  const int d4 = dh * 512 + threadIdx.x * 4;
  const float* cb = cur + ((size_t)b * S_) * D_;
  const float* qb = q + (size_t)b * S_;

  v4f acc = {0.f, 0.f, 0.f, 0.f};
  for (int i = 0; i < CS; ++i) {
    const int s = j * CS + i;
    const float qs = qb[s];
    v4f x = *(const v4f*)(cb + (size_t)s * D_ + d4);
    acc += qs * x;
  }
  *(v4f*)(G + ((size_t)(b * NC + j)) * D_ + d4) = acc;
}

// ---------------------------------------------------------------------------
// Kernel 4: in-place exclusive scan of G across the NC chunks, per (b,d).
// Loads all NC values first (pipelined), scans in registers, stores back.
// ---------------------------------------------------------------------------
__global__ __launch_bounds__(256) void k4_scan(float* __restrict__ G) {
  const int col = blockIdx.x * 256 + threadIdx.x;   // 0..B_*D_-1
  const int b = col >> 10;
  const int d = col & (D_ - 1);
  float* gp = G + (size_t)b * NC * D_ + d;
  float v[NC];
#pragma unroll
  for (int j = 0; j < NC; ++j) v[j] = gp[(size_t)j * D_];
  float run = 0.f;
#pragma unroll
  for (int j = 0; j < NC; ++j) { float x = v[j]; v[j] = run; run += x; }
#pragma unroll
  for (int j = 0; j < NC; ++j) gp[(size_t)j * D_] = v[j];
}

// ---------------------------------------------------------------------------
// Kernel 5: out[b,s,:] = sum_l p[s,l]*V_col[l,b,s,:] + invZ_s * prefix(q*cur)
// Running in-chunk scan seeded by the exclusive chunk prefix in G.
// History stream (268MB, read-once) uses non-temporal loads; NT stores out.
// ---------------------------------------------------------------------------
__global__ __launch_bounds__(128) void k5_out(const float* __restrict__ hist,
                                              const float* __restrict__ cur,
                                              const float* __restrict__ sc,
                                              const float* __restrict__ q,
                                              const float* __restrict__ invZ,
                                              const float* __restrict__ G,
                                              float* __restrict__ out) {
  const int blk = blockIdx.x;
  const int dh = blk & 1;
  const int j  = (blk >> 1) & (NC - 1);
  const int b  = blk >> 7;
  const int d4 = dh * 512 + threadIdx.x * 4;

  v4f R = *(const v4f*)(G + ((size_t)(b * NC + j)) * D_ + d4);
  const size_t bsD = (size_t)b * S_ * D_;
  const size_t planeSz = (size_t)B_ * S_ * D_;

  for (int i = 0; i < CS; ++i) {
    const int s = j * CS + i;
    const size_t rowOff = bsD + (size_t)s * D_ + d4;
    const float* pv = sc + ((size_t)(b * S_ + s)) * LP1;
    const float qs = q[(size_t)b * S_ + s];
    const float iz = invZ[(size_t)b * S_ + s];

    v4f x = *(const v4f*)(cur + rowOff);
    R += qs * x;                                   // running causal scan
    v4f acc = pv[L_] * x;                          // layer 8 == current
#pragma unroll
    for (int l = 0; l < L_; ++l) {
      v4f v = __builtin_nontemporal_load((const v4f*)(hist + (size_t)l * planeSz + rowOff));
      acc += pv[l] * v;
    }
    v4f o = acc + R * iz;
    __builtin_nontemporal_store(o, (v4f*)(out + rowOff));
  }
}

// ---------------------------------------------------------------------------
extern "C" void kernel_launch(void* const* d_in, const int* in_sizes, int n_in,
                              void* d_out, int out_size, void* d_ws, size_t ws_size,
                              hipStream_t stream) {
  (void)in_sizes; (void)n_in; (void)out_size; (void)ws_size;
  const float* hist  = (const float*)d_in[0];   // (L,B,S,D)
  const float* cur   = (const float*)d_in[1];   // (B,S,D)
  const float* w     = (const float*)d_in[2];   // (D,)
  const float* gamma = (const float*)d_in[3];   // (D,)
  const float* beta  = (const float*)d_in[4];   // (D,)
  float* out = (float*)d_out;

  // workspace layout (floats); all offsets multiple of 4 floats (16B aligned)
  float* ws   = (float*)d_ws;
  float* wg   = ws;                           // D_            = 1024
  float* scal = ws + 1024;                    // 2 (+6 pad)
  float* sc   = ws + 1032;                    // B*S*LP1       = 73728
  float* qv   = sc + (size_t)B_ * S_ * LP1;   // B*S           = 8192
  float* invZ = qv + (size_t)B_ * S_;         // B*S           = 8192
  float* G    = invZ + (size_t)B_ * S_;       // B*NC*D        = 262144

  k0_prep   <<<1,   1024, 0, stream>>>(w, gamma, beta, wg, scal);
  k1_scores <<<576,  256, 0, stream>>>(hist, cur, wg, scal, sc);   // 4608 waves
  k2_softmax<<<B_,  1024, 0, stream>>>(sc, qv, invZ);
  k3_chunks <<<B_ * NC * 2, 128, 0, stream>>>(cur, qv, G);
  k4_scan   <<<(B_ * D_) / 256, 256, 0, stream>>>(G);
  k5_out    <<<B_ * NC * 2, 128, 0, stream>>>(hist, cur, sc, qv, invZ, G, out);
}